// HashEncoding_24223615549953
// MI455X (gfx1250) — compile-verified
//
#include <hip/hip_runtime.h>
#include <math.h>

// ---------------------------------------------------------------------------
// Instant-NGP HashEncoding forward for MI455X (gfx1250).
//  - 20 scales, grid sizes = round(geomspace(16, 8192, 20)):
//      dense (g^3 <= 2^21): 16,22,31,43,59,83,115   (scales 0..6)
//      hashed (T = 2^21):   159,221,307,427,592,823,1142,1586,2203,3059,4248,5899,8192
//  - per point: 8-corner trilerp per scale; out[N, 40] = concat * 10
//  - scales 0,1 tables staged into LDS via gfx1250 async global->LDS copies
//  - output stores are non-temporal (protect L2 residency of the 16MB tables)
// ---------------------------------------------------------------------------

#define BLOCK 512

typedef float v2f __attribute__((ext_vector_type(2)));
typedef float v4f __attribute__((ext_vector_type(4)));
typedef int   v4i __attribute__((ext_vector_type(4)));

static constexpr int   NUM_SCALES = 20;
static constexpr int   NUM_DENSE  = 7;           // g^3 <= 2^21
static constexpr unsigned HASH_MASK = (1u << 21) - 1u;
static constexpr unsigned PI2 = 19349663u, PI3 = 83492791u;
static constexpr float WORLD_MIN  = -2.0f;
static constexpr float INV_EXTENT = 0.25f;       // 1/(WORLD_MAX-WORLD_MIN)
static constexpr float PRECOND    = 10.0f;

__device__ static constexpr int GRIDS[NUM_SCALES] = {
    16, 22, 31, 43, 59, 83, 115, 159, 221, 307,
    427, 592, 823, 1142, 1586, 2203, 3059, 4248, 5899, 8192};

// LDS-staged dense tables (scales 0 and 1)
static constexpr int LDS0 = 16 * 16 * 16 * 2;    //  8192 floats = 32 KB
static constexpr int LDS1 = 22 * 22 * 22 * 2;    // 21296 floats = 85184 B
static constexpr int LDS_TOTAL = LDS0 + LDS1;    // 117952 B  (<= 320 KB/WGP)

struct TablePtrs { const float* t[NUM_SCALES]; };

#if defined(__HIP_DEVICE_COMPILE__) && defined(__gfx1250__) && \
    __has_builtin(__builtin_amdgcn_global_load_async_to_lds_b128)
#define USE_ASYNC_LDS 1
#else
#define USE_ASYNC_LDS 0
#endif

__device__ __forceinline__ void stage_table(const float* __restrict__ src,
                                            float* __restrict__ dst_lds,
                                            int nfloats, int tid) {
#if USE_ASYNC_LDS
  // Builtin signature (from clang diagnostic): (v4i AS1*, v4i AS3*, Ii, Ii)
  typedef __attribute__((address_space(1))) v4i* gptr_t;
  typedef __attribute__((address_space(3))) v4i* lptr_t;
  for (int o = tid * 4; o < nfloats; o += BLOCK * 4) {
    __builtin_amdgcn_global_load_async_to_lds_b128(
        (gptr_t)(src + o), (lptr_t)(dst_lds + o), 0, 0);
  }
#else
  for (int o = tid * 4; o < nfloats; o += BLOCK * 4) {
    const v4f v = *reinterpret_cast<const v4f*>(src + o);
    *reinterpret_cast<v4f*>(dst_lds + o) = v;
  }
#endif
}

__global__ void __launch_bounds__(BLOCK)
hash_encoding_kernel(const float* __restrict__ x, TablePtrs tp,
                     float* __restrict__ out, int n) {
  __shared__ float lds_tab[LDS_TOTAL];

  const int tid = threadIdx.x;
  // Stage the two coarsest dense grids into LDS (async DMA path on gfx1250).
  stage_table(tp.t[0], lds_tab, LDS0, tid);
  stage_table(tp.t[1], lds_tab + LDS0, LDS1, tid);
#if USE_ASYNC_LDS
  asm volatile("s_wait_asynccnt 0" ::: "memory");
#endif
  __syncthreads();

  const int i = blockIdx.x * BLOCK + tid;
  if (i >= n) return;

  const float* xp = x + 3 * (size_t)i;
  const float nx = (xp[0] - WORLD_MIN) * INV_EXTENT;
  const float ny = (xp[1] - WORLD_MIN) * INV_EXTENT;
  const float nz = (xp[2] - WORLD_MIN) * INV_EXTENT;

  v2f* __restrict__ outv = reinterpret_cast<v2f*>(out) + (size_t)i * NUM_SCALES;

#pragma unroll
  for (int s = 0; s < NUM_SCALES; ++s) {
    const int g = GRIDS[s];
    const float lx = nx * (float)g - 0.5f;
    const float ly = ny * (float)g - 0.5f;
    const float lz = nz * (float)g - 0.5f;
    const float fx = floorf(lx), fy = floorf(ly), fz = floorf(lz);
    const float w1x = lx - fx, w1y = ly - fy, w1z = lz - fz;
    const float wx[2] = {1.0f - w1x, w1x};
    const float wy[2] = {1.0f - w1y, w1y};
    const float wz[2] = {1.0f - w1z, w1z};
    const int ix = (int)fx, iy = (int)fy, iz = (int)fz;

    float a0 = 0.0f, a1 = 0.0f;

    if (s < NUM_DENSE) {
      // Dense grid, CONSTANT_OUTSIDE (zero) padding: skip out-of-range corners.
      const float* base = (s == 0) ? lds_tab
                        : (s == 1) ? (lds_tab + LDS0)
                                   : tp.t[s];
      const v2f* t = reinterpret_cast<const v2f*>(base);
#pragma unroll
      for (int c = 0; c < 8; ++c) {
        const int ox = (c >> 2) & 1, oy = (c >> 1) & 1, oz = c & 1;
        const int jx = ix + ox, jy = iy + oy, jz = iz + oz;
        const float w = wx[ox] * wy[oy] * wz[oz];
        if ((unsigned)jx < (unsigned)g && (unsigned)jy < (unsigned)g &&
            (unsigned)jz < (unsigned)g) {
          const v2f v = t[(jx * g + jy) * g + jz];
          a0 = fmaf(w, v.x, a0);
          a1 = fmaf(w, v.y, a1);
        }
      }
    } else {
      // XOR hash path, T = 2^21 (power of two -> mask).
      const v2f* t = reinterpret_cast<const v2f*>(tp.t[s]);
      const unsigned hx[2] = {(unsigned)ix, (unsigned)(ix + 1)};
      const unsigned hy[2] = {(unsigned)iy * PI2, (unsigned)(iy + 1) * PI2};
      const unsigned hz[2] = {(unsigned)iz * PI3, (unsigned)(iz + 1) * PI3};
#pragma unroll
      for (int c = 0; c < 8; ++c) {
        const int ox = (c >> 2) & 1, oy = (c >> 1) & 1, oz = c & 1;
        const float w = wx[ox] * wy[oy] * wz[oz];
        const unsigned idx = (hx[ox] ^ hy[oy] ^ hz[oz]) & HASH_MASK;
        const v2f v = t[idx];
        a0 = fmaf(w, v.x, a0);
        a1 = fmaf(w, v.y, a1);
      }
    }

    v2f r;
    r.x = a0 * PRECOND;
    r.y = a1 * PRECOND;
    // Streaming output (160 MB): non-temporal so it doesn't evict hot tables.
    __builtin_nontemporal_store(r, &outv[s]);
  }
}

extern "C" void kernel_launch(void* const* d_in, const int* in_sizes, int n_in,
                              void* d_out, int out_size, void* d_ws, size_t ws_size,
                              hipStream_t stream) {
  (void)n_in; (void)out_size; (void)d_ws; (void)ws_size;
  const float* x = (const float*)d_in[0];
  TablePtrs tp;
  for (int s = 0; s < NUM_SCALES; ++s) tp.t[s] = (const float*)d_in[1 + s];
  const int n = in_sizes[0] / 3;
  const int grid = (n + BLOCK - 1) / BLOCK;
  hash_encoding_kernel<<<grid, BLOCK, 0, stream>>>(x, tp, (float*)d_out, n);
}